// Virtual_KD_73461120631550
// MI455X (gfx1250) — compile-verified
//
#include <hip/hip_runtime.h>
#include <cmath>

// -------------------------------------------------------------------------
// Virtual-teacher KD loss, MI455X (gfx1250).
// Memory-bound row reduction (824MB @ 23.3TB/s ~= 35us floor): stream each
// 196KB row HBM -> LDS with GLOBAL_LOAD_ASYNC_TO_LDS_B128 (ASYNCcnt-tracked,
// 3-buffer / 2-ahead per-wave pipeline, non-temporal), online logsumexp +
// sum accumulation, deterministic two-stage reduction.
// -------------------------------------------------------------------------

#define TPB            256
#define WAVES          8
#define NBUF           3                         // LDS buffers per wave
#define STAGE_PER_WAVE 512                       // floats per wave per stage
#define STAGE_FLOATS   (STAGE_PER_WAVE * WAVES)  // 4096 floats per block stage
#define INSTR_PER_STG  4                         // async b128 instrs / wave / stage
#define EXP2C          0.72134752044448170f      // log2(e)/T, T = 2

__device__ __forceinline__ float fexp2(float x) { return __builtin_amdgcn_exp2f(x); }

// online-softmax accumulate of one element: S = sum 2^((x-M)*log2e/T)
__device__ __forceinline__ void kd_acc(float v, float& m, float& s, float& sx) {
    sx += v;
    float nm = fmaxf(m, v);
    s = s * fexp2((m - nm) * EXP2C) + fexp2((v - nm) * EXP2C);
    m = nm;
}

// combine two online-softmax partials
__device__ __forceinline__ void kd_merge(float om, float os, float ox,
                                         float& m, float& s, float& sx) {
    float nm = fmaxf(m, om);
    s = s * fexp2((m - nm) * EXP2C) + os * fexp2((om - nm) * EXP2C);
    m = nm;
    sx += ox;
}

__global__ void __launch_bounds__(TPB)
kd_row_kernel(const int* __restrict__ targets, const float* __restrict__ logits,
              float* __restrict__ rowloss, int K, double p_t, double p_o)
{
    __shared__ __align__(16) float sbuf[WAVES * NBUF * STAGE_PER_WAVE]; // 48 KB
    __shared__ float red_m[WAVES], red_s[WAVES], red_x[WAVES];

    const int tid  = threadIdx.x;
    const int w    = tid >> 5;
    const int lane = tid & 31;
    const int b    = blockIdx.x;
    const float* row = logits + (size_t)b * (size_t)K;

    // Target-class logit (only thread 0's copy is used).
    float xt = 0.0f;
    if (tid == 0) xt = row[targets[b]];

    // Rows are K*4 = 201028 bytes -> not 16B aligned for odd b. Peel a head
    // of 0..3 floats so the async B128 bulk is 16B aligned.
    uint32_t addr_lo = (uint32_t)(uintptr_t)row;
    int h = (int)(((16u - (addr_lo & 15u)) & 15u) >> 2);
    const float* arow = row + h;
    int S = (K - h) / STAGE_FLOATS;              // full async stages (12 here)

    float m = -INFINITY, s = 0.0f, sx = 0.0f;

    // addrspace(3) offsets live in the low 32 bits of the generic pointer.
    uint32_t sbase = (uint32_t)(uintptr_t)(&sbuf[0]);

    // Issue one stage for this wave: 4x async 128b NT loads = 2KB, ASYNCcnt += 4.
    auto issue = [&](int stage, int buf) {
        uint32_t loff = sbase + (uint32_t)((w * NBUF + buf) * (STAGE_PER_WAVE * 4))
                              + (uint32_t)(lane * 16);
        uint32_t g0 = (uint32_t)((stage * STAGE_FLOATS + w * STAGE_PER_WAVE
                                  + lane * 4) * 4);
        #pragma unroll
        for (int j = 0; j < INSTR_PER_STG; ++j) {
            asm volatile("global_load_async_to_lds_b128 %0, %1, %2 th:TH_LOAD_NT"
                         :: "v"(loff + (uint32_t)(j * 512)),
                            "v"(g0   + (uint32_t)(j * 512)),
                            "s"(arow)
                         : "memory");
        }
    };

    // Prologue: fill the pipeline two stages deep.
    if (S > 0) issue(0, 0);
    if (S > 1) issue(1, 1);

    for (int st = 0; st < S; ++st) {
        int rem = S - 1 - st;                    // stages after this one
        if (rem >= 2) {
            // WAR: DS reads and async-LDS writes are mutually unordered;
            // drain our LDS reads before overwriting buffer (st+2)%3.
            asm volatile("s_wait_dscnt 0" ::: "memory");
            issue(st + 2, (st + 2) % NBUF);
            asm volatile("s_wait_asynccnt 8" ::: "memory"); // stage st landed
        } else if (rem == 1) {
            asm volatile("s_wait_asynccnt 4" ::: "memory");
        } else {
            asm volatile("s_wait_asynccnt 0" ::: "memory");
        }
        const float4* rp =
            (const float4*)(&sbuf[(w * NBUF + (st % NBUF)) * STAGE_PER_WAVE]);
        #pragma unroll
        for (int j = 0; j < 4; ++j) {
            float4 v = rp[lane + 32 * j];
            kd_acc(v.x, m, s, sx); kd_acc(v.y, m, s, sx);
            kd_acc(v.z, m, s, sx); kd_acc(v.w, m, s, sx);
        }
    }

    // Head (alignment peel) + tail (K - h - S*4096 leftover), direct loads.
    if (tid < h) kd_acc(row[tid], m, s, sx);
    for (int i = h + S * STAGE_FLOATS + tid; i < K; i += TPB)
        kd_acc(row[i], m, s, sx);

    // Wave32 butterfly reduce of (m, s, sx) with online-softmax combine.
    #pragma unroll
    for (int off = 16; off >= 1; off >>= 1) {
        float om = __shfl_xor(m,  off, 32);
        float os = __shfl_xor(s,  off, 32);
        float ox = __shfl_xor(sx, off, 32);
        kd_merge(om, os, ox, m, s, sx);
    }
    if (lane == 0) { red_m[w] = m; red_s[w] = s; red_x[w] = sx; }
    __syncthreads();

    if (tid == 0) {
        float M = red_m[0], Se = red_s[0], SX = red_x[0];
        #pragma unroll
        for (int i = 1; i < WAVES; ++i)
            kd_merge(red_m[i], red_s[i], red_x[i], M, Se, SX);
        // lse = logsumexp(x/T); final math in double (once per row).
        double lse     = (double)M * 0.5 + log((double)Se);
        double lsm_t   = (double)xt * 0.5 - lse;
        double sum_lsm = (double)SX * 0.5 - (double)K * lse;
        double loss    = -(p_t * lsm_t + p_o * (sum_lsm - lsm_t));
        rowloss[b] = (float)loss;
    }
}

// Deterministic fixed-order mean of B row losses.
__global__ void __launch_bounds__(TPB)
kd_mean_kernel(const float* __restrict__ rows, float* __restrict__ out, int n)
{
    __shared__ double sm[TPB];
    double acc = 0.0;
    for (int i = threadIdx.x; i < n; i += TPB) acc += (double)rows[i];
    sm[threadIdx.x] = acc;
    __syncthreads();
    for (int s = TPB / 2; s > 0; s >>= 1) {
        if (threadIdx.x < s) sm[threadIdx.x] += sm[threadIdx.x + s];
        __syncthreads();
    }
    if (threadIdx.x == 0) out[0] = (float)(sm[0] / (double)n);
}

extern "C" void kernel_launch(void* const* d_in, const int* in_sizes, int n_in,
                              void* d_out, int out_size, void* d_ws, size_t ws_size,
                              hipStream_t stream) {
    const int*   targets = (const int*)d_in[0];
    const float* logits  = (const float*)d_in[1];
    float* rowloss = (float*)d_ws;          // B floats of scratch
    float* outf    = (float*)d_out;

    const int B = in_sizes[0];
    const int K = in_sizes[1] / B;

    // Virtual teacher softmax at temperature T: only two distinct logit
    // values -> two row-independent probabilities, computed in double here.
    const double T = 2.0, COR = 0.9;
    double base  = (1.0 - COR) / ((double)K - 1.0);
    double ea    = exp(COR / T);
    double eb    = exp(base / T);
    double denom = ea + ((double)K - 1.0) * eb;
    double p_t   = ea / denom;
    double p_o   = eb / denom;

    kd_row_kernel<<<B, TPB, 0, stream>>>(targets, logits, rowloss, K, p_t, p_o);
    kd_mean_kernel<<<1, TPB, 0, stream>>>(rowloss, outf, B);
}